// HammingDistanceAttention_44650480009373
// MI455X (gfx1250) — compile-verified
//
#include <hip/hip_runtime.h>
#include <hip/hip_bf16.h>

typedef float v2f __attribute__((ext_vector_type(2)));
typedef float v4f __attribute__((ext_vector_type(4)));
typedef float v8f __attribute__((ext_vector_type(8)));

#define N_SEQ      2048
#define DHEAD      64
#define ROWS       16
#define LDS_STRIDE 2052   // 2048 + 4 floats pad to break 64-bank periodicity

__global__ __launch_bounds__(256)
void hamming_attn_kernel(const float* __restrict__ q,
                         const float* __restrict__ k,
                         float* __restrict__ out)
{
    extern __shared__ float smem[];
    float* scores = smem;                        // ROWS * LDS_STRIDE floats
    float* redmax = smem + ROWS * LDS_STRIDE;    // 16 rows x 16 partials
    float* redsum = redmax + 256;                // 16 rows x 16 partials

    const int tid  = threadIdx.x;
    const int lane = tid & 31;
    const int wid  = tid >> 5;

    const int bh      = blockIdx.x >> 7;         // 128 row tiles per (b,h)
    const int rowTile = blockIdx.x & 127;

    const float* qbase = q + ((size_t)bh * N_SEQ + (size_t)rowTile * ROWS) * DHEAD;
    const float* kbase = k + (size_t)bh * N_SEQ * DHEAD;

    // ---- A operand: qa = 2q-1, 16x64, kept in registers for the whole block row.
    // V_WMMA_F32_16X16X4_F32 A layout (16x4 per step):
    //   lanes 0-15 : M = lane,    VGPR0 = K0, VGPR1 = K1
    //   lanes 16-31: M = lane-16, VGPR0 = K2, VGPR1 = K3
    const int mrow = lane & 15;
    const int koff = (lane >> 4) * 2;
    v2f a2[16];
    {
        const float* ap = qbase + mrow * DHEAD + koff;
#pragma unroll
        for (int kk = 0; kk < 16; ++kk) {
            float x0 = ap[kk * 4 + 0];
            float x1 = ap[kk * 4 + 1];
            a2[kk][0] = 2.0f * x0 - 1.0f;
            a2[kk][1] = 2.0f * x1 - 1.0f;
        }
    }

    // ---- each wave computes 16 of the 128 column tiles (stride 8)
#pragma unroll 1
    for (int t = 0; t < 16; ++t) {
        const int ct = wid + t * 8;
        const float* bp = kbase + (size_t)(ct * 16 + mrow) * DHEAD + koff;
        v2f b2[16];
#pragma unroll
        for (int kk = 0; kk < 16; ++kk) {
            float x0 = bp[kk * 4 + 0];
            float x1 = bp[kk * 4 + 1];
            b2[kk][0] = 2.0f * x0 - 1.0f;
            b2[kk][1] = 2.0f * x1 - 1.0f;
        }

        v8f c = {};
#pragma unroll
        for (int kk = 0; kk < 16; ++kk) {
            // D = A(16x4,f32) x B(4x16,f32) + C(16x16,f32)
            c = __builtin_amdgcn_wmma_f32_16x16x4_f32(
                    /*neg_a=*/false, a2[kk],
                    /*neg_b=*/false, b2[kk],
                    /*c_mod=*/(short)0, c,
                    /*reuse_a=*/false, /*reuse_b=*/false);
        }

        // C/D layout: lane holds column n = (lane&15); VGPR r holds row r (+8 for lanes 16-31)
        const int col   = ct * 16 + mrow;
        const int rbase = (lane >> 4) * 8;
#pragma unroll
        for (int r = 0; r < 8; ++r) {
            scores[(rbase + r) * LDS_STRIDE + col] = 0.5f + c[r] * (1.0f / 128.0f);
        }
    }
    __syncthreads();

    // ---- row softmax over the 16 x 2048 LDS tile: 16 threads per row
    const int row = tid >> 4;
    const int sub = tid & 15;
    float* srow = scores + row * LDS_STRIDE;

    // pass 1: row max
    float m = -3.402823466e+38f;
#pragma unroll 4
    for (int j = 0; j < 32; ++j) {
        const v4f v = *(const v4f*)(srow + j * 64 + sub * 4);
        m = fmaxf(m, fmaxf(fmaxf(v.x, v.y), fmaxf(v.z, v.w)));
    }
    redmax[row * 16 + sub] = m;
    __syncthreads();
    float rowmax = redmax[row * 16];
#pragma unroll
    for (int i = 1; i < 16; ++i) rowmax = fmaxf(rowmax, redmax[row * 16 + i]);

    // pass 2: exp in place + row sum
    float s = 0.0f;
#pragma unroll 4
    for (int j = 0; j < 32; ++j) {
        v4f* p = (v4f*)(srow + j * 64 + sub * 4);
        v4f v = *p;
        v.x = __expf(v.x - rowmax);
        v.y = __expf(v.y - rowmax);
        v.z = __expf(v.z - rowmax);
        v.w = __expf(v.w - rowmax);
        s += (v.x + v.y) + (v.z + v.w);
        *p = v;
    }
    redsum[row * 16 + sub] = s;
    __syncthreads();
    float rowsum = 0.0f;
#pragma unroll
    for (int i = 0; i < 16; ++i) rowsum += redsum[row * 16 + i];
    const float inv = 1.0f / rowsum;

    // pass 3: normalize + non-temporal streaming store (output is write-once,
    // never re-read: NT hint keeps the 1.07 GB stream from thrashing L2,
    // preserving K-panel residency for the other 127 blocks per (b,h)).
    float* orow = out + ((size_t)bh * N_SEQ + (size_t)rowTile * ROWS + row) * N_SEQ;
#pragma unroll 4
    for (int j = 0; j < 32; ++j) {
        const int c0 = j * 64 + sub * 4;
        v4f v = *(const v4f*)(srow + c0);
        v *= inv;
        __builtin_nontemporal_store(v, (v4f*)(orow + c0));
    }
}

extern "C" void kernel_launch(void* const* d_in, const int* in_sizes, int n_in,
                              void* d_out, int out_size, void* d_ws, size_t ws_size,
                              hipStream_t stream) {
    (void)in_sizes; (void)n_in; (void)out_size; (void)d_ws; (void)ws_size;
    const float* q = (const float*)d_in[0];
    const float* k = (const float*)d_in[1];
    float* out = (float*)d_out;

    const int B = 8, H = 8;
    const dim3 grid(B * H * (N_SEQ / ROWS));   // 8192 blocks
    const dim3 block(256);                      // 8 wave32s
    const size_t shmem = (size_t)(ROWS * LDS_STRIDE + 512) * sizeof(float); // ~130 KB

    hamming_attn_kernel<<<grid, block, shmem, stream>>>(q, k, out);
}